// L12ProxSolver_88399016886712
// MI455X (gfx1250) — compile-verified
//
#include <hip/hip_runtime.h>

typedef __attribute__((ext_vector_type(2))) float v2f;
typedef __attribute__((ext_vector_type(8))) float v8f;

#define BS 256
#define M  128
#define N  512
#define P  64
#define NT 576   // N + P
#define AUG_STRIDE 257

// A = [G | W], row-major per batch: A[r][c], r<128, c<576
__device__ __forceinline__ float loadA(const float* __restrict__ G,
                                       const float* __restrict__ W,
                                       int b, int r, int c) {
    return (c < N) ? G[((size_t)b * M + r) * N + c]
                   : W[((size_t)b * M + r) * P + (c - N)];
}

// ---------------------------------------------------------------------------
// Kernel 1: S = A * A^T per batch via V_WMMA_F32_16X16X4_F32.
// One wave per 16x16 tile; grid = (8*8 tiles, bs).
// A-frag (16x4 f32, 2 VGPR): lane half h, l=lane&15: row = 16*ti + l,
//   K = k0 + 2h + {0,1}. B = A^T has the same per-lane load pattern.
// D (16x16 f32, 8 VGPR): VGPR g, lane half h -> row = g + 8h, col = lane&15.
// ---------------------------------------------------------------------------
__global__ void __launch_bounds__(32)
gram_kernel(const float* __restrict__ G, const float* __restrict__ W,
            float* __restrict__ S) {
    const int b    = blockIdx.y;
    const int ti   = blockIdx.x >> 3;
    const int tj   = blockIdx.x & 7;
    const int lane = threadIdx.x;
    const int half = lane >> 4;
    const int l15  = lane & 15;

    const int rowA = ti * 16 + l15;
    const int rowB = tj * 16 + l15;

    v8f acc = {};
    for (int k0 = 0; k0 < NT; k0 += 4) {
        const int k = k0 + 2 * half;      // 512 is 4-aligned: never straddles G/W
        v2f a, bb;
        a.x  = loadA(G, W, b, rowA, k);
        a.y  = loadA(G, W, b, rowA, k + 1);
        bb.x = loadA(G, W, b, rowB, k);
        bb.y = loadA(G, W, b, rowB, k + 1);
        acc = __builtin_amdgcn_wmma_f32_16x16x4_f32(false, a, false, bb,
                                                    (short)0, acc, false, false);
    }

    float* Sb = S + (size_t)b * M * M;
    #pragma unroll
    for (int g = 0; g < 8; ++g)
        Sb[(ti * 16 + g + 8 * half) * M + (tj * 16 + l15)] = acc[g];
}

// ---------------------------------------------------------------------------
// Kernel 2: Minv = S^{-1} per batch. Gauss-Jordan on [S | I] in dynamic LDS
// (SPD -> no pivoting). 256 threads: thread t owns row t>>1, col half (t&1).
// LDS: 128 * 257 * 4 = 131,584 B (< 320 KB WGP LDS).
// ---------------------------------------------------------------------------
__global__ void __launch_bounds__(256)
invert_kernel(const float* __restrict__ S, float* __restrict__ Minv) {
    extern __shared__ float aug[];      // [128][257], cols 0..255 used
    const int b = blockIdx.x;
    const int t = threadIdx.x;
    const float* Sb = S + (size_t)b * M * M;

    for (int idx = t; idx < M * 2 * M; idx += 256) {
        const int r = idx >> 8;
        const int c = idx & 255;
        aug[r * AUG_STRIDE + c] = (c < M) ? Sb[r * M + c]
                                          : ((c - M == r) ? 1.0f : 0.0f);
    }
    __syncthreads();

    const int rr = t >> 1;
    const int c0 = (t & 1) * M;
    for (int k = 0; k < M; ++k) {
        const float pinv = 1.0f / aug[k * AUG_STRIDE + k];
        __syncthreads();                          // all pinv reads before scale write
        aug[k * AUG_STRIDE + t] *= pinv;          // scale pivot row (256 cols)
        __syncthreads();
        const float f = aug[rr * AUG_STRIDE + k]; // capture eliminator pre-write
        __syncthreads();
        if (rr != k) {
            float*       rowr = &aug[rr * AUG_STRIDE + c0];
            const float* rowk = &aug[k  * AUG_STRIDE + c0];
            #pragma unroll 8
            for (int c = 0; c < M; ++c)
                rowr[c] = fmaf(-f, rowk[c], rowr[c]);
        }
        __syncthreads();
    }

    float* Mb = Minv + (size_t)b * M * M;
    for (int idx = t; idx < M * M; idx += 256) {
        const int r = idx >> 7;
        const int c = idx & 127;
        Mb[r * M + c] = aug[r * AUG_STRIDE + (M + c)];
    }
}

// ---------------------------------------------------------------------------
// Kernel 3: C = Minv * A per batch (128 x 576, K = 128) via WMMA f32.
// One wave per 16x16 output tile; grid = (8 * 36 tiles, bs).
// A-frag rows come from Minv; B-frag is A itself (rows = K, cols = output col).
// Column tile is entirely in G (tj < 32) or entirely in W (tj >= 32) since
// 512 = 32*16, so the G/W select is block-uniform.
// ---------------------------------------------------------------------------
__global__ void __launch_bounds__(32)
projmat_kernel(const float* __restrict__ G, const float* __restrict__ W,
               const float* __restrict__ Minv, float* __restrict__ C) {
    const int b    = blockIdx.y;
    const int ti   = blockIdx.x / 36;   // 0..7   (output rows)
    const int tj   = blockIdx.x % 36;   // 0..35  (output cols)
    const int lane = threadIdx.x;
    const int half = lane >> 4;
    const int l15  = lane & 15;

    const float* Mb  = Minv + (size_t)b * M * M;
    const int    rowA = ti * 16 + l15;  // Minv row
    const int    colB = tj * 16 + l15;  // A column

    v8f acc = {};
    for (int k0 = 0; k0 < M; k0 += 4) {
        const int k = k0 + 2 * half;
        v2f a, bb;
        a.x  = Mb[rowA * M + k];
        a.y  = Mb[rowA * M + k + 1];
        bb.x = loadA(G, W, b, k,     colB);
        bb.y = loadA(G, W, b, k + 1, colB);
        acc = __builtin_amdgcn_wmma_f32_16x16x4_f32(false, a, false, bb,
                                                    (short)0, acc, false, false);
    }

    float* Cb = C + (size_t)b * M * NT;
    #pragma unroll
    for (int g = 0; g < 8; ++g)
        Cb[(ti * 16 + g + 8 * half) * NT + (tj * 16 + l15)] = acc[g];
}

// ---------------------------------------------------------------------------
// Kernel 4: 30-iteration solver. One 128-thread block per batch.
//   sh = soft(xi); v = [2sh - xi ; T]; u = C v; temp = v - A^T u;
//   z = temp[:512]; tau = temp[512:]; xi += z - sh.   (eta is dead.)
// xi/sh/z in registers; v/u in LDS. u-pass uses float4 (b128) row loads;
// A^T pass is lane-coalesced scalar column loads.
// ---------------------------------------------------------------------------
__global__ void __launch_bounds__(128)
solver_kernel(const float* __restrict__ G, const float* __restrict__ W,
              const float* __restrict__ T, const float* __restrict__ D1,
              const float* __restrict__ D2, const float* __restrict__ C,
              const float* __restrict__ alpha_p, const int* __restrict__ iters_p,
              float* __restrict__ out) {
    __shared__ __align__(16) float v[NT];
    __shared__ float u[M];

    const int   b     = blockIdx.x;
    const int   t     = threadIdx.x;    // 0..127
    const float alpha = *alpha_p;
    const int   iters = *iters_p;

    float aj[4], rdn[4], xi[4], sh[4], zr[4];
    #pragma unroll
    for (int q = 0; q < 4; ++q) {
        const int j  = t + 128 * q;
        const float d1 = D1[(size_t)b * N + j];
        const float d2 = D2[(size_t)b * N + j];
        aj[q]  = alpha * fabsf(d1);
        rdn[q] = 1.0f / (1.0f + 2.0f * alpha * d2 * d2);
        xi[q]  = 0.0f;
        sh[q]  = 0.0f;
        zr[q]  = 0.0f;
    }
    float taur = 0.0f;
    if (t < P) v[N + t] = T[(size_t)b * P + t];   // tail of v is constant = T

    for (int it = 0; it < iters; ++it) {
        // sh = soft_threshold(xi); v[:512] = 2*sh - xi
        #pragma unroll
        for (int q = 0; q < 4; ++q) {
            const float x  = xi[q];
            const float xm = x - aj[q];
            const float xp = x + aj[q];
            const float s  = (xm > 0.0f) ? xm * rdn[q]
                           : ((xp < 0.0f) ? xp * rdn[q] : 0.0f);
            sh[q] = s;
            v[t + 128 * q] = 2.0f * s - x;
        }
        __syncthreads();

        // u = C v  (thread t = row t; float4 row loads, v broadcast from LDS)
        {
            const float4* Cr = (const float4*)(C + ((size_t)b * M + t) * NT);
            const float4* v4 = (const float4*)v;
            float acc = 0.0f;
            #pragma unroll 4
            for (int i = 0; i < NT / 4; ++i) {
                const float4 cv = Cr[i];
                const float4 vv = v4[i];
                acc = fmaf(cv.x, vv.x, acc);
                acc = fmaf(cv.y, vv.y, acc);
                acc = fmaf(cv.z, vv.z, acc);
                acc = fmaf(cv.w, vv.w, acc);
            }
            u[t] = acc;
        }
        __syncthreads();

        // temp[c] = v[c] - sum_r A[r][c] * u[r]   (lane-coalesced columns)
        #pragma unroll
        for (int q = 0; q < 5; ++q) {
            const int c = t + 128 * q;
            if (c < NT) {
                const float* Ac;
                int stride;
                if (c < N) { Ac = G + (size_t)b * M * N + c;       stride = N; }
                else       { Ac = W + (size_t)b * M * P + (c - N); stride = P; }
                float acc = v[c];
                for (int r = 0; r < M; ++r)
                    acc = fmaf(-Ac[(size_t)r * stride], u[r], acc);
                if (c < N) {                 // z part: update xi
                    zr[q] = acc;
                    xi[q] = xi[q] + acc - sh[q];
                } else {                     // tau part (threads 0..63, q==4)
                    taur = acc;
                }
            }
        }
        __syncthreads();   // protect v[] rewrite next iteration
    }

    #pragma unroll
    for (int q = 0; q < 4; ++q)
        out[(size_t)b * N + t + 128 * q] = zr[q];
    if (t < P)
        out[(size_t)BS * N + (size_t)b * P + t] = taur;
}

// ---------------------------------------------------------------------------
// Launch. Workspace: S (16 MB) + Minv (16 MB) + C (75.5 MB) ~= 108 MB.
// Inputs: G(0) W(1) T(2) D1(3) D2(4) bs(5) alpha(6) max_iters(7)
// ---------------------------------------------------------------------------
extern "C" void kernel_launch(void* const* d_in, const int* in_sizes, int n_in,
                              void* d_out, int out_size, void* d_ws, size_t ws_size,
                              hipStream_t stream) {
    const float* G     = (const float*)d_in[0];
    const float* W     = (const float*)d_in[1];
    const float* T     = (const float*)d_in[2];
    const float* D1    = (const float*)d_in[3];
    const float* D2    = (const float*)d_in[4];
    const float* alpha = (const float*)d_in[6];
    const int*   iters = (const int*)d_in[7];

    float* S    = (float*)d_ws;
    float* Minv = S    + (size_t)BS * M * M;
    float* C    = Minv + (size_t)BS * M * M;

    gram_kernel<<<dim3(64, BS), 32, 0, stream>>>(G, W, S);
    invert_kernel<<<dim3(BS), 256, AUG_STRIDE * M * sizeof(float), stream>>>(S, Minv);
    projmat_kernel<<<dim3(288, BS), 32, 0, stream>>>(G, W, Minv, C);
    solver_kernel<<<dim3(BS), 128, 0, stream>>>(G, W, T, D1, D2, C, alpha, iters,
                                                (float*)d_out);
}